// GraphAE_85315230367791
// MI455X (gfx1250) — compile-verified
//
#include <hip/hip_runtime.h>

// GraphSAGE autoencoder for MI455X (gfx1250, wave32).
// Memory-bound (~1.5 GB HBM traffic vs ~11 GFLOP) -> keep f32, use
// V_WMMA_F32_16X16X4_F32 for all dense GEMMs, minimize gather bytes by
// projecting h to 64 dims BEFORE the layer-2 aggregation (linearity of mean).

#define NN 50000
#define NE 800000
#define IN_DIM 128
#define HIDDEN 256
#define LATENT 64

typedef __attribute__((ext_vector_type(2))) float v2f;
typedef __attribute__((ext_vector_type(8))) float v8f;

__global__ void zero_kernel(float* __restrict__ p, int n) {
  int i = blockIdx.x * blockDim.x + threadIdx.x;
  if (i < n) p[i] = 0.0f;
}

__global__ void count_kernel(const int* __restrict__ dst, float* __restrict__ cnt) {
  int e = blockIdx.x * blockDim.x + threadIdx.x;
  if (e < NE) atomicAdd(&cnt[dst[e]], 1.0f);
}

// scatter-add D-dim features along edges; each thread handles 4 consecutive floats
template <int D, int LOG4>  // LOG4 = log2(D/4)
__global__ void scatter_kernel(const int* __restrict__ src, const int* __restrict__ dst,
                               const float* __restrict__ feat, float* __restrict__ acc) {
  int i = blockIdx.x * blockDim.x + threadIdx.x;   // over NE * D/4
  int e  = i >> LOG4;
  int k4 = (i & ((D / 4) - 1)) * 4;
  int s = src[e], d = dst[e];
  const float4 v = *(const float4*)(feat + s * D + k4);
  float* out = acc + d * D + k4;
  atomicAdd(out + 0, v.x);
  atomicAdd(out + 1, v.y);
  atomicAdd(out + 2, v.z);
  atomicAdd(out + 3, v.w);
}

// h = relu((sum1 * inv_cnt) @ W1l + x @ W1r + b1)   [NN,HIDDEN], K=IN_DIM
__global__ void gemm1_kernel(const float* __restrict__ x, const float* __restrict__ sum1,
                             const float* __restrict__ cnt,
                             const float* __restrict__ W1l, const float* __restrict__ W1r,
                             const float* __restrict__ b1, float* __restrict__ h) {
  int tid = blockIdx.x * blockDim.x + threadIdx.x;
  int w = tid >> 5, lane = tid & 31, half = lane >> 4, r = lane & 15;
  const int NT = HIDDEN / 16;                 // 16 tiles along output
  int m0 = (w / NT) * 16, n0 = (w % NT) * 16; // waves = 3125*16 (exact grid)
  int arow = m0 + r;
  float inv = 1.0f / fmaxf(cnt[arow], 1.0f);
  v8f acc = {};
  for (int k = 0; k < IN_DIM; k += 4) {
    int ka = k + 2 * half;
    v2f aM = *(const v2f*)(sum1 + arow * IN_DIM + ka);
    aM.x *= inv; aM.y *= inv;
    v2f aX = *(const v2f*)(x + arow * IN_DIM + ka);
    v2f bL, bR;
    bL.x = W1l[(ka + 0) * HIDDEN + n0 + r];
    bL.y = W1l[(ka + 1) * HIDDEN + n0 + r];
    bR.x = W1r[(ka + 0) * HIDDEN + n0 + r];
    bR.y = W1r[(ka + 1) * HIDDEN + n0 + r];
    acc = __builtin_amdgcn_wmma_f32_16x16x4_f32(false, aM, false, bL, (short)0, acc, false, false);
    acc = __builtin_amdgcn_wmma_f32_16x16x4_f32(false, aX, false, bR, (short)0, acc, false, false);
  }
  float bias = b1[n0 + r];
#pragma unroll
  for (int v = 0; v < 8; v++) {
    int mrow = m0 + v + 8 * half;
    h[mrow * HIDDEN + n0 + r] = fmaxf(acc[v] + bias, 0.0f);
  }
}

// p = h @ W2l   [NN,LATENT], K=HIDDEN
__global__ void gemmp_kernel(const float* __restrict__ h, const float* __restrict__ W2l,
                             float* __restrict__ p, int nwaves) {
  int tid = blockIdx.x * blockDim.x + threadIdx.x;
  int w = tid >> 5, lane = tid & 31, half = lane >> 4, r = lane & 15;
  if (w >= nwaves) return;                       // wave-uniform guard
  const int NT = LATENT / 16;                    // 4
  int m0 = (w / NT) * 16, n0 = (w % NT) * 16;
  int arow = m0 + r;
  v8f acc = {};
  for (int k = 0; k < HIDDEN; k += 4) {
    int ka = k + 2 * half;
    v2f a = *(const v2f*)(h + arow * HIDDEN + ka);
    v2f b;
    b.x = W2l[(ka + 0) * LATENT + n0 + r];
    b.y = W2l[(ka + 1) * LATENT + n0 + r];
    acc = __builtin_amdgcn_wmma_f32_16x16x4_f32(false, a, false, b, (short)0, acc, false, false);
  }
#pragma unroll
  for (int v = 0; v < 8; v++) {
    int mrow = m0 + v + 8 * half;
    p[mrow * LATENT + n0 + r] = acc[v];
  }
}

// z = h @ W2r + sump * inv_cnt + b2   [NN,LATENT], K=HIDDEN
__global__ void gemmz_kernel(const float* __restrict__ h, const float* __restrict__ W2r,
                             const float* __restrict__ sump, const float* __restrict__ cnt,
                             const float* __restrict__ b2, float* __restrict__ z, int nwaves) {
  int tid = blockIdx.x * blockDim.x + threadIdx.x;
  int w = tid >> 5, lane = tid & 31, half = lane >> 4, r = lane & 15;
  if (w >= nwaves) return;
  const int NT = LATENT / 16;
  int m0 = (w / NT) * 16, n0 = (w % NT) * 16;
  int arow = m0 + r;
  v8f acc = {};
  for (int k = 0; k < HIDDEN; k += 4) {
    int ka = k + 2 * half;
    v2f a = *(const v2f*)(h + arow * HIDDEN + ka);
    v2f b;
    b.x = W2r[(ka + 0) * LATENT + n0 + r];
    b.y = W2r[(ka + 1) * LATENT + n0 + r];
    acc = __builtin_amdgcn_wmma_f32_16x16x4_f32(false, a, false, b, (short)0, acc, false, false);
  }
  float bias = b2[n0 + r];
#pragma unroll
  for (int v = 0; v < 8; v++) {
    int mrow = m0 + v + 8 * half;
    float inv = 1.0f / fmaxf(cnt[mrow], 1.0f);
    z[mrow * LATENT + n0 + r] = acc[v] + sump[mrow * LATENT + n0 + r] * inv + bias;
  }
}

// x_hat = z @ Wd + bd   [NN,IN_DIM], K=LATENT
__global__ void gemmd_kernel(const float* __restrict__ z, const float* __restrict__ Wd,
                             const float* __restrict__ bd, float* __restrict__ xhat) {
  int tid = blockIdx.x * blockDim.x + threadIdx.x;
  int w = tid >> 5, lane = tid & 31, half = lane >> 4, r = lane & 15;
  const int NT = IN_DIM / 16;                    // 8, waves = 3125*8 (exact grid)
  int m0 = (w / NT) * 16, n0 = (w % NT) * 16;
  int arow = m0 + r;
  v8f acc = {};
  for (int k = 0; k < LATENT; k += 4) {
    int ka = k + 2 * half;
    v2f a = *(const v2f*)(z + arow * LATENT + ka);
    v2f b;
    b.x = Wd[(ka + 0) * IN_DIM + n0 + r];
    b.y = Wd[(ka + 1) * IN_DIM + n0 + r];
    acc = __builtin_amdgcn_wmma_f32_16x16x4_f32(false, a, false, b, (short)0, acc, false, false);
  }
  float bias = bd[n0 + r];
#pragma unroll
  for (int v = 0; v < 8; v++) {
    int mrow = m0 + v + 8 * half;
    xhat[mrow * IN_DIM + n0 + r] = acc[v] + bias;
  }
}

extern "C" void kernel_launch(void* const* d_in, const int* in_sizes, int n_in,
                              void* d_out, int out_size, void* d_ws, size_t ws_size,
                              hipStream_t stream) {
  const float* x   = (const float*)d_in[0];
  const int*   ei  = (const int*)d_in[1];   // [2,E] int32: row0=src, row1=dst
  const float* W1l = (const float*)d_in[2];
  const float* b1  = (const float*)d_in[3];
  const float* W1r = (const float*)d_in[4];
  const float* W2l = (const float*)d_in[5];
  const float* b2  = (const float*)d_in[6];
  const float* W2r = (const float*)d_in[7];
  const float* Wd  = (const float*)d_in[8];
  const float* bd  = (const float*)d_in[9];
  const int* src = ei;
  const int* dst = ei + NE;

  float* z    = (float*)d_out;                 // [NN,64]
  float* xhat = z + (size_t)NN * LATENT;       // [NN,128]

  // workspace layout: [sum1 | cnt | sump]  (zeroed prefix) | h | p
  float* sum1 = (float*)d_ws;                  // NN*128
  float* cnt  = sum1 + (size_t)NN * IN_DIM;    // NN
  float* sump = cnt + NN;                      // NN*64
  float* h    = sump + (size_t)NN * LATENT;    // NN*256
  float* p    = h + (size_t)NN * HIDDEN;       // NN*64
  (void)in_sizes; (void)n_in; (void)out_size; (void)ws_size;

  const int zeroN = NN * IN_DIM + NN + NN * LATENT;   // 9,650,000
  zero_kernel<<<(zeroN + 255) / 256, 256, 0, stream>>>(sum1, zeroN);

  count_kernel<<<(NE + 255) / 256, 256, 0, stream>>>(dst, cnt);

  // layer-1 aggregation of x (128 dims): NE * 32 threads
  scatter_kernel<IN_DIM, 5><<<NE * 32 / 256, 256, 0, stream>>>(src, dst, x, sum1);

  // h = relu(mean1 @ W1l + x @ W1r + b1): 3125*16 waves = 6250 blocks (exact)
  gemm1_kernel<<<(NN / 16) * (HIDDEN / 16) * 32 / 256, 256, 0, stream>>>(
      x, sum1, cnt, W1l, W1r, b1, h);

  // p = h @ W2l: 3125*4 = 12500 waves
  {
    int nwaves = (NN / 16) * (LATENT / 16);
    gemmp_kernel<<<(nwaves * 32 + 255) / 256, 256, 0, stream>>>(h, W2l, p, nwaves);
  }

  // layer-2 aggregation of p (64 dims): NE * 16 threads
  scatter_kernel<LATENT, 4><<<NE * 16 / 256, 256, 0, stream>>>(src, dst, p, sump);

  // z = h @ W2r + mean(p) + b2
  {
    int nwaves = (NN / 16) * (LATENT / 16);
    gemmz_kernel<<<(nwaves * 32 + 255) / 256, 256, 0, stream>>>(h, W2r, sump, cnt, b2, z, nwaves);
  }

  // x_hat = z @ Wd + bd: 3125*8 waves = 3125 blocks (exact)
  gemmd_kernel<<<(NN / 16) * (IN_DIM / 16) * 32 / 256, 256, 0, stream>>>(z, Wd, bd, xhat);
}